// RayTransformer_79061757985005
// MI455X (gfx1250) — compile-verified
//
#include <hip/hip_runtime.h>
#include <math.h>

// ---------------------------------------------------------------------------
// Problem constants (from reference)
// ---------------------------------------------------------------------------
#define NPTS   32768      // RN*SN = 512*64
#define RNUM   512
#define SNUM   64
#define NVIEW  8
#define DV     48         // d_view = IMG_FEAT_DIM + FEA_VOL_DIM
#define DRAY   56         // d_ray  = d_view + PE_D_HID
#define LV     9          // tokens per point (view_token + 8 views)
#define LRAY   64         // samples per ray

typedef __attribute__((ext_vector_type(2))) float v2f;
typedef __attribute__((ext_vector_type(8))) float v8f;

// ---------------------------------------------------------------------------
// Generic fp32 WMMA GEMM:  C[M,N] = act(A[M,K] @ W[N,K]^T + bias[N])
// V_WMMA_F32_16X16X4_F32: one wave computes a 16x16 C tile, K steps of 4.
// Layouts per CDNA5 ISA 7.12.2:
//   A 16x4 : all lanes hold rows M=lane%16; lanes<16 K={kk,kk+1}, lanes>=16 K={kk+2,kk+3}
//   B 4x16 : all lanes hold cols N=lane%16; same K split
//   C 16x16: vgpr r, lanes<16 -> row r, lanes>=16 -> row r+8; col = lane%16
//
// REQUIREMENT: M % 16 == 0 (true for every launch in this file) -> no row guards.
// Fast path (wave-uniform): full N tile, K%4==0, even lda/ldw -> unconditional
// 8-byte v2f loads + 2-stage software pipeline so loads overlap the WMMA.
// Slow path (edge tiles): loads are UNCONDITIONAL at clamped addresses; OOB
// contributions are zeroed via value selects (v_cndmask) -> no EXEC churn.
// ---------------------------------------------------------------------------
__global__ void gemm_wmma_f32(const float* __restrict__ A, int lda,
                              const float* __restrict__ W, int ldw,
                              const float* __restrict__ bias,
                              float* __restrict__ C, int ldc,
                              int M, int N, int K, int relu)
{
    const int wave = threadIdx.x >> 5;
    const int lane = threadIdx.x & 31;
    const int m0 = (blockIdx.y * (blockDim.x >> 5) + wave) * 16;
    const int n0 = blockIdx.x * 16;
    if (m0 >= M) return;                 // wave-uniform exit: EXEC stays all-1s

    const int l16 = lane & 15;
    const int hl  = lane >> 4;
    const int arow = m0 + l16;           // always valid: M % 16 == 0
    const int bcol = n0 + l16;

    v8f acc = {0.f, 0.f, 0.f, 0.f, 0.f, 0.f, 0.f, 0.f};

    // wave-uniform path select (n0, K, lda, ldw are uniform)
    const bool fast = (n0 + 16 <= N) && ((K & 3) == 0) && (((lda | ldw) & 1) == 0);

    if (fast) {
        const float* Ap = A + (size_t)arow * lda + hl * 2;
        const float* Wp = W + (size_t)bcol * ldw + hl * 2;
        v2f a0 = *(const v2f*)Ap;        // global_load_b64, no predication
        v2f b0 = *(const v2f*)Wp;
        for (int kk = 4; kk < K; kk += 4) {
            Ap += 4;
            Wp += 4;
            const v2f a1 = *(const v2f*)Ap;   // prefetch next K-step
            const v2f b1 = *(const v2f*)Wp;
            acc = __builtin_amdgcn_wmma_f32_16x16x4_f32(
                      false, a0, false, b0, (short)0, acc, false, false);
            a0 = a1;
            b0 = b1;
        }
        acc = __builtin_amdgcn_wmma_f32_16x16x4_f32(
                  false, a0, false, b0, (short)0, acc, false, false);

        const float bb = (bias != nullptr) ? bias[bcol] : 0.f;
#pragma unroll
        for (int r = 0; r < 8; ++r) {
            float val = acc[r] + bb;
            if (relu) val = fmaxf(val, 0.f);
            C[(size_t)(m0 + r + hl * 8) * ldc + bcol] = val;
        }
    } else {
        const bool bv = bcol < N;
        const int  bcolc = bv ? bcol : (N - 1);          // clamped, always legal
        const float* Ar = A + (size_t)arow * lda;
        const float* Wr = W + (size_t)bcolc * ldw;
        for (int kk = 0; kk < K; kk += 4) {
            const int k0 = kk + hl * 2;
            const bool k0v = k0 < K;
            const bool k1v = (k0 + 1) < K;
            const int k0c = k0v ? k0 : (K - 1);          // clamped indices
            const int k1c = k1v ? (k0 + 1) : (K - 1);
            // unconditional loads at clamped addresses
            const float a0 = Ar[k0c];
            const float a1 = Ar[k1c];
            const float b0 = Wr[k0c];
            const float b1 = Wr[k1c];
            v2f a, b;
            a.x = k0v ? a0 : 0.f;                        // value selects only
            a.y = k1v ? a1 : 0.f;
            b.x = (bv && k0v) ? b0 : 0.f;
            b.y = (bv && k1v) ? b1 : 0.f;
            acc = __builtin_amdgcn_wmma_f32_16x16x4_f32(
                      false, a, false, b, (short)0, acc, false, false);
        }
        const float bb = (bias != nullptr && bv) ? bias[bcol] : 0.f;
#pragma unroll
        for (int r = 0; r < 8; ++r) {
            if (bv) {
                float val = acc[r] + bb;
                if (relu) val = fmaxf(val, 0.f);
                C[(size_t)(m0 + r + hl * 8) * ldc + bcol] = val;
            }
        }
    }
}

// ---------------------------------------------------------------------------
// Linear attention (per sequence, per head).  KV is DHxDH register-resident.
//   KV = sum_s (elu(k)+1)^T (v/L) ; Z = 1/(Q.Ksum + 1e-6) ; msg = Q KV * Z * L
// msg overwrites the q buffer in place.
// ---------------------------------------------------------------------------
__device__ __forceinline__ float elu1(float x) { return x > 0.f ? x + 1.f : expf(x); }

template <int DH, int L>
__global__ void lin_attn(float* __restrict__ q, const float* __restrict__ k,
                         const float* __restrict__ v, int nseq, int D)
{
    const int NH = D / DH;
    const int t = blockIdx.x * blockDim.x + threadIdx.x;
    if (t >= nseq * NH) return;
    const int n = t / NH, h = t % NH;
    const float invL = 1.f / (float)L;

    float KV[DH][DH];
    float Ks[DH];
#pragma unroll
    for (int d = 0; d < DH; ++d) {
        Ks[d] = 0.f;
#pragma unroll
        for (int e = 0; e < DH; ++e) KV[d][e] = 0.f;
    }
    for (int s = 0; s < L; ++s) {
        const size_t base = ((size_t)n * L + s) * D + h * DH;
        float kp[DH], vv[DH];
#pragma unroll
        for (int d = 0; d < DH; ++d) { kp[d] = elu1(k[base + d]); vv[d] = v[base + d] * invL; }
#pragma unroll
        for (int d = 0; d < DH; ++d) {
            Ks[d] += kp[d];
#pragma unroll
            for (int e = 0; e < DH; ++e) KV[d][e] += kp[d] * vv[e];
        }
    }
    for (int l = 0; l < L; ++l) {
        const size_t base = ((size_t)n * L + l) * D + h * DH;
        float qp[DH];
        float zden = 1e-6f;
#pragma unroll
        for (int d = 0; d < DH; ++d) { qp[d] = elu1(q[base + d]); zden += qp[d] * Ks[d]; }
        const float scale = (float)L / zden;
#pragma unroll
        for (int e = 0; e < DH; ++e) {
            float m = 0.f;
#pragma unroll
            for (int d = 0; d < DH; ++d) m += qp[d] * KV[d][e];
            q[base + e] = m * scale;
        }
    }
}

// ---------------------------------------------------------------------------
// cat = [x | LN(msgm)]     (one thread per row)
// ---------------------------------------------------------------------------
__global__ void ln_concat_kernel(const float* __restrict__ x, const float* __restrict__ m,
                                 const float* __restrict__ g, const float* __restrict__ b,
                                 float* __restrict__ cat, int rows, int D)
{
    const int r = blockIdx.x * blockDim.x + threadIdx.x;
    if (r >= rows) return;
    const float* mr = m + (size_t)r * D;
    float mean = 0.f;
    for (int d = 0; d < D; ++d) mean += mr[d];
    mean /= (float)D;
    float var = 0.f;
    for (int d = 0; d < D; ++d) { const float t = mr[d] - mean; var += t * t; }
    var /= (float)D;
    const float inv = rsqrtf(var + 1e-5f);
    const float* xr = x + (size_t)r * D;
    float* cr = cat + (size_t)r * 2 * D;
    for (int d = 0; d < D; ++d) cr[d] = xr[d];
    for (int d = 0; d < D; ++d) cr[D + d] = (mr[d] - mean) * inv * g[d] + b[d];
}

// x += LN(m2)   (residual, in place on token buffer)
__global__ void ln_residual_kernel(float* __restrict__ x, const float* __restrict__ m,
                                   const float* __restrict__ g, const float* __restrict__ b,
                                   int rows, int D)
{
    const int r = blockIdx.x * blockDim.x + threadIdx.x;
    if (r >= rows) return;
    const float* mr = m + (size_t)r * D;
    float mean = 0.f;
    for (int d = 0; d < D; ++d) mean += mr[d];
    mean /= (float)D;
    float var = 0.f;
    for (int d = 0; d < D; ++d) { const float t = mr[d] - mean; var += t * t; }
    var /= (float)D;
    const float inv = rsqrtf(var + 1e-5f);
    float* xr = x + (size_t)r * D;
    for (int d = 0; d < D; ++d) xr[d] += (mr[d] - mean) * inv * g[d] + b[d];
}

// ---------------------------------------------------------------------------
// Prep 1: trilinear volume sample -> channels 32..47 of all 8 view tokens,
// and broadcast view_token into token 0.  One thread per point.
// ---------------------------------------------------------------------------
__global__ void vol_token_kernel(const float* __restrict__ p3d, const float* __restrict__ vol,
                                 const float* __restrict__ vtok, float* __restrict__ tokens)
{
    const int pt = blockIdx.x * blockDim.x + threadIdx.x;
    if (pt >= NPTS) return;
    const float x = p3d[pt * 3 + 0], y = p3d[pt * 3 + 1], z = p3d[pt * 3 + 2];
    const float ix = (x + 1.f) * 0.5f * 63.f;
    const float iy = (y + 1.f) * 0.5f * 63.f;
    const float iz = (z + 1.f) * 0.5f * 63.f;
    const float x0 = floorf(ix), y0 = floorf(iy), z0 = floorf(iz);
    const float wx = ix - x0, wy = iy - y0, wz = iz - z0;

    int   idx8[8];
    float w8[8];
    int c = 0;
    for (int dz = 0; dz < 2; ++dz)
        for (int dy = 0; dy < 2; ++dy)
            for (int dx = 0; dx < 2; ++dx) {
                const int xi = (int)fminf(fmaxf(x0 + (float)dx, 0.f), 63.f);
                const int yi = (int)fminf(fmaxf(y0 + (float)dy, 0.f), 63.f);
                const int zi = (int)fminf(fmaxf(z0 + (float)dz, 0.f), 63.f);
                idx8[c] = (zi * 64 + yi) * 64 + xi;
                w8[c] = (dx ? wx : 1.f - wx) * (dy ? wy : 1.f - wy) * (dz ? wz : 1.f - wz);
                ++c;
            }
    float* t0 = tokens + (size_t)pt * LV * DV;
    for (int ch = 0; ch < DV; ++ch) t0[ch] = vtok[ch];
    for (int ch = 0; ch < 16; ++ch) {
        const float* vc = vol + (size_t)ch * 262144;
        float val = 0.f;
#pragma unroll
        for (int i = 0; i < 8; ++i) val += w8[i] * vc[idx8[i]];
        for (int v = 0; v < NVIEW; ++v)
            tokens[((size_t)pt * LV + 1 + v) * DV + 32 + ch] = val;
    }
}

// ---------------------------------------------------------------------------
// Prep 2: per (point, view): dir_relative, projection (pip output), mask,
// bilinear sample of image features (32ch -> token) and rgb (3ch).
// ---------------------------------------------------------------------------
__global__ void view_sample_kernel(const float* __restrict__ p3d,
                                   const float* __restrict__ imgs,
                                   const float* __restrict__ feat,
                                   const float* __restrict__ refpi,
                                   const float* __restrict__ spi,
                                   const float* __restrict__ sp,
                                   float* __restrict__ tokens,
                                   float* __restrict__ dirrel,
                                   float* __restrict__ maskb,
                                   float* __restrict__ rgbb,
                                   float* __restrict__ pip_out)
{
    const int t = blockIdx.x * blockDim.x + threadIdx.x;
    if (t >= NPTS * NVIEW) return;
    const int pt = t >> 3, v = t & 7;
    const float x = p3d[pt * 3 + 0], y = p3d[pt * 3 + 1], z = p3d[pt * 3 + 2];

    // dir_relative = normalize(p - ref_cam) - normalize(p - src_cam)
    {
        const float r1x = x - refpi[3], r1y = y - refpi[7], r1z = z - refpi[11];
        const float n1 = rsqrtf(r1x * r1x + r1y * r1y + r1z * r1z);
        const float r2x = x - spi[v * 16 + 3], r2y = y - spi[v * 16 + 7], r2z = z - spi[v * 16 + 11];
        const float n2 = rsqrtf(r2x * r2x + r2y * r2y + r2z * r2z);
        dirrel[(size_t)t * 3 + 0] = r1x * n1 - r2x * n2;
        dirrel[(size_t)t * 3 + 1] = r1y * n1 - r2y * n2;
        dirrel[(size_t)t * 3 + 2] = r1z * n1 - r2z * n2;
    }

    // projection
    const float* P = sp + v * 16;
    const float hx = P[0] * x + P[1] * y + P[2]  * z + P[3];
    const float hy = P[4] * x + P[5] * y + P[6]  * z + P[7];
    const float hz = P[8] * x + P[9] * y + P[10] * z + P[11];
    const float mvd = (hz > 0.f) ? 1.f : 0.f;
    const float gx = hx / hz, gy = hy / hz;
    pip_out[((size_t)v * 2 + 0) * NPTS + pt] = gx;
    pip_out[((size_t)v * 2 + 1) * NPTS + pt] = gy;
    const float inb = (gx >= 0.f && gx <= 255.f && gy >= 0.f && gy <= 255.f) ? 1.f : 0.f;
    maskb[t] = inb * mvd;

    // bilinear corner weights / offsets (shared by all channels)
    const float x0 = floorf(gx), y0 = floorf(gy);
    const float wx = gx - x0, wy = gy - y0;
    int   off4[4];
    float w4[4];
    int c = 0;
    for (int dy = 0; dy < 2; ++dy)
        for (int dx = 0; dx < 2; ++dx) {
            const float cx = x0 + (float)dx, cy = y0 + (float)dy;
            const float valid = (cx >= 0.f && cx <= 255.f && cy >= 0.f && cy <= 255.f) ? 1.f : 0.f;
            const int xi = (int)fminf(fmaxf(cx, 0.f), 255.f);
            const int yi = (int)fminf(fmaxf(cy, 0.f), 255.f);
            off4[c] = yi * 256 + xi;
            w4[c] = valid * ((dx ? wx : 1.f - wx) * (dy ? wy : 1.f - wy));
            ++c;
        }

    float* tokrow = tokens + ((size_t)pt * LV + 1 + v) * DV;
    for (int ch = 0; ch < 32; ++ch) {
        const float* fp = feat + ((size_t)v * 32 + ch) * 65536;
        tokrow[ch] = w4[0] * fp[off4[0]] + w4[1] * fp[off4[1]] +
                     w4[2] * fp[off4[2]] + w4[3] * fp[off4[3]];
    }
    for (int ch = 0; ch < 3; ++ch) {
        const float* ip = imgs + ((size_t)v * 3 + ch) * 65536;
        rgbb[(size_t)t * 3 + ch] = w4[0] * ip[off4[0]] + w4[1] * ip[off4[1]] +
                                   w4[2] * ip[off4[2]] + w4[3] * ip[off4[3]];
    }
}

// ---------------------------------------------------------------------------
// Ray tokens: xr[row] = [token0(row) | order_posenc(8, s=row%64)]
// ---------------------------------------------------------------------------
__global__ void ray_token_kernel(const float* __restrict__ tokens, float* __restrict__ xr)
{
    const int row = blockIdx.x * blockDim.x + threadIdx.x;
    if (row >= NPTS) return;
    const int s = row & 63;
    const float* t0 = tokens + (size_t)row * LV * DV;
    float* dst = xr + (size_t)row * DRAY;
    for (int ch = 0; ch < DV; ++ch) dst[ch] = t0[ch];
#pragma unroll
    for (int j = 0; j < 8; ++j) {
        const float ex = (float)(2 * (j >> 1)) / 8.f;
        const float ang = (float)s * powf(10000.f, -ex);
        dst[DV + j] = (j & 1) ? cosf(ang) : sinf(ang);
    }
}

// ---------------------------------------------------------------------------
// Radiance MLP input rows (chunked): [view_feature(48) | dir_relative(3) | pad]
// ---------------------------------------------------------------------------
__global__ void rad_in_kernel(const float* __restrict__ tokens, const float* __restrict__ dirrel,
                              float* __restrict__ rin, int rowBase, int nrows)
{
    const int r = blockIdx.x * blockDim.x + threadIdx.x;
    if (r >= nrows) return;
    const int g = rowBase + r;         // global (pt, v) row index
    const int pt = g >> 3, v = g & 7;
    const float* src = tokens + ((size_t)pt * LV + 1 + v) * DV;
    float* dst = rin + (size_t)r * 52;
    for (int ch = 0; ch < DV; ++ch) dst[ch] = src[ch];
    dst[48] = dirrel[(size_t)g * 3 + 0];
    dst[49] = dirrel[(size_t)g * 3 + 1];
    dst[50] = dirrel[(size_t)g * 3 + 2];
    dst[51] = 0.f;
}

// ---------------------------------------------------------------------------
// Masked softmax over views + rgb blend -> radiance output
// ---------------------------------------------------------------------------
__global__ void blend_kernel(const float* __restrict__ xw, const float* __restrict__ maskb,
                             const float* __restrict__ rgbb, float* __restrict__ outr,
                             int ptBase, int nchunk)
{
    const int i = blockIdx.x * blockDim.x + threadIdx.x;
    if (i >= nchunk) return;
    const int pt = ptBase + i;
    float a[NVIEW];
    float mx = -INFINITY;
#pragma unroll
    for (int v = 0; v < NVIEW; ++v) {
        const float m = maskb[(size_t)pt * NVIEW + v];
        const float val = (m == 0.f) ? -1e9f : xw[(size_t)i * NVIEW + v];
        a[v] = val;
        mx = fmaxf(mx, val);
    }
    float ssum = 0.f;
#pragma unroll
    for (int v = 0; v < NVIEW; ++v) { a[v] = expf(a[v] - mx); ssum += a[v]; }
    const float inv = 1.f / ssum;
    float r = 0.f, g = 0.f, b = 0.f;
#pragma unroll
    for (int v = 0; v < NVIEW; ++v) {
        const float w = a[v] * inv;
        const float* rp = rgbb + ((size_t)pt * NVIEW + v) * 3;
        r += w * rp[0];
        g += w * rp[1];
        b += w * rp[2];
    }
    outr[(size_t)pt * 3 + 0] = r;
    outr[(size_t)pt * 3 + 1] = g;
    outr[(size_t)pt * 3 + 2] = b;
}

// ---------------------------------------------------------------------------
// Host helpers
// ---------------------------------------------------------------------------
static inline void launch_gemm(hipStream_t s, const float* A, int lda, const float* W, int ldw,
                               const float* bias, float* C, int ldc, int M, int N, int K, int relu)
{
    dim3 blk(128);                       // 4 waves per block, 1 wave = 16x16 tile
    dim3 grd((N + 15) / 16, (M + 63) / 64);
    gemm_wmma_f32<<<grd, blk, 0, s>>>(A, lda, W, ldw, bias, C, ldc, M, N, K, relu);
}

struct EncW { const float *q, *k, *v, *m, *w1, *w2, *g1, *b1, *g2, *b2; };

static void run_encoder(hipStream_t s, float* X, int nseq, int L, int D, int dh,
                        const EncW& w, float* qb, float* kb, float* vb, float* cat, float* h1)
{
    const int rows = nseq * L;
    launch_gemm(s, X, D, w.q, D, nullptr, qb, D, rows, D, D, 0);
    launch_gemm(s, X, D, w.k, D, nullptr, kb, D, rows, D, D, 0);
    launch_gemm(s, X, D, w.v, D, nullptr, vb, D, rows, D, D, 0);
    const int nh = nseq * 8;
    const int ablk = (nh + 255) / 256;
    if (dh == 6) lin_attn<6, LV><<<ablk, 256, 0, s>>>(qb, kb, vb, nseq, D);
    else         lin_attn<7, LRAY><<<ablk, 256, 0, s>>>(qb, kb, vb, nseq, D);
    launch_gemm(s, qb, D, w.m, D, nullptr, kb, D, rows, D, D, 0);   // msgm -> kb
    const int rblk = (rows + 255) / 256;
    ln_concat_kernel<<<rblk, 256, 0, s>>>(X, kb, w.g1, w.b1, cat, rows, D);
    launch_gemm(s, cat, 2 * D, w.w1, 2 * D, nullptr, h1, 2 * D, rows, 2 * D, 2 * D, 1);
    launch_gemm(s, h1, 2 * D, w.w2, 2 * D, nullptr, vb, D, rows, D, 2 * D, 0); // m2 -> vb
    ln_residual_kernel<<<rblk, 256, 0, s>>>(X, vb, w.g2, w.b2, rows, D);
}

// ---------------------------------------------------------------------------
// Input index map (assumes recursive insertion-order flattening of
// setup_inputs(): tensors first, then params dict depth-first in insertion
// order; each (W, b) tuple contributes W then b).
// ---------------------------------------------------------------------------
enum {
    IN_P3D = 0, IN_IMGS, IN_REFPI, IN_SPI, IN_SP, IN_FEAT, IN_VOL,
    IN_VTOK,
    IN_VT_Q, IN_VT_K, IN_VT_V, IN_VT_M, IN_VT_W1, IN_VT_W2,
    IN_VT_G1, IN_VT_B1, IN_VT_G2, IN_VT_B2,
    IN_RT_Q, IN_RT_K, IN_RT_V, IN_RT_M, IN_RT_W1, IN_RT_W2,
    IN_RT_G1, IN_RT_B1, IN_RT_G2, IN_RT_B2,
    IN_D_W0, IN_D_B0, IN_D_W1, IN_D_B1, IN_D_W2, IN_D_B2,
    IN_R_W0, IN_R_B0, IN_R_W1, IN_R_B1, IN_R_W2, IN_R_B2
};

extern "C" void kernel_launch(void* const* d_in, const int* in_sizes, int n_in,
                              void* d_out, int out_size, void* d_ws, size_t ws_size,
                              hipStream_t stream)
{
    (void)in_sizes; (void)n_in; (void)out_size;

    // ---- workspace layout (floats) ----
    const size_t OFF_TOKENS = 0;                               // 294912 x 48
    const size_t OFF_DIRREL = OFF_TOKENS + (size_t)NPTS * LV * DV;   // 262144 x 3
    const size_t OFF_MASK   = OFF_DIRREL + (size_t)NPTS * NVIEW * 3; // 262144
    const size_t OFF_RGB    = OFF_MASK + (size_t)NPTS * NVIEW;       // 262144 x 3
    const size_t OFF_XR     = OFF_RGB + (size_t)NPTS * NVIEW * 3;    // 32768 x 56
    const size_t OFF_POOL   = OFF_XR + (size_t)NPTS * DRAY;
    const size_t POOL_FLOATS = 12845056;   // max stage need (rt encoder scratch)
    const size_t NEED = (OFF_POOL + POOL_FLOATS) * sizeof(float);
    if (ws_size < NEED) return;            // not enough scratch: bail safely

    float* ws = (float*)d_ws;
    float* tokens = ws + OFF_TOKENS;
    float* dirrel = ws + OFF_DIRREL;
    float* maskb  = ws + OFF_MASK;
    float* rgbb   = ws + OFF_RGB;
    float* xr     = ws + OFF_XR;
    float* pool   = ws + OFF_POOL;

    float* out_rad  = (float*)d_out;            // 98304
    float* out_srdf = out_rad + 98304;          // 32768
    float* out_pip  = out_srdf + 32768;         // 524288

    const float* p3d   = (const float*)d_in[IN_P3D];
    const float* imgs  = (const float*)d_in[IN_IMGS];
    const float* refpi = (const float*)d_in[IN_REFPI];
    const float* spi   = (const float*)d_in[IN_SPI];
    const float* sp    = (const float*)d_in[IN_SP];
    const float* feat  = (const float*)d_in[IN_FEAT];
    const float* vol   = (const float*)d_in[IN_VOL];
    const float* vtok  = (const float*)d_in[IN_VTOK];

    EncW vt = { (const float*)d_in[IN_VT_Q], (const float*)d_in[IN_VT_K],
                (const float*)d_in[IN_VT_V], (const float*)d_in[IN_VT_M],
                (const float*)d_in[IN_VT_W1], (const float*)d_in[IN_VT_W2],
                (const float*)d_in[IN_VT_G1], (const float*)d_in[IN_VT_B1],
                (const float*)d_in[IN_VT_G2], (const float*)d_in[IN_VT_B2] };
    EncW rt = { (const float*)d_in[IN_RT_Q], (const float*)d_in[IN_RT_K],
                (const float*)d_in[IN_RT_V], (const float*)d_in[IN_RT_M],
                (const float*)d_in[IN_RT_W1], (const float*)d_in[IN_RT_W2],
                (const float*)d_in[IN_RT_G1], (const float*)d_in[IN_RT_B1],
                (const float*)d_in[IN_RT_G2], (const float*)d_in[IN_RT_B2] };

    // ---- stage 1: gather/prep (fills token buffer, dir_rel, mask, rgb, pip) ----
    vol_token_kernel<<<NPTS / 256, 256, 0, stream>>>(p3d, vol, vtok, tokens);
    view_sample_kernel<<<(NPTS * NVIEW) / 256, 256, 0, stream>>>(
        p3d, imgs, feat, refpi, spi, sp, tokens, dirrel, maskb, rgbb, out_pip);

    // ---- stage 2: view-token encoder (N=32768 seqs, L=9, D=48), chunked ----
    {
        const int CH = 4096;                 // sequences per chunk
        const int rows = CH * LV;            // 36864
        float* qb  = pool;
        float* kb  = qb + (size_t)rows * DV;
        float* vb  = kb + (size_t)rows * DV;
        float* cat = vb + (size_t)rows * DV;
        float* h1  = cat + (size_t)rows * 2 * DV;
        for (int c = 0; c < NPTS / CH; ++c) {
            float* X = tokens + (size_t)c * CH * LV * DV;
            run_encoder(stream, X, CH, LV, DV, 6, vt, qb, kb, vb, cat, h1);
        }
    }

    // ---- stage 3: ray encoder (N=512 rays, L=64, D=56) ----
    ray_token_kernel<<<NPTS / 256, 256, 0, stream>>>(tokens, xr);
    {
        const int rows = NPTS;               // 512 * 64
        float* qb  = pool;
        float* kb  = qb + (size_t)rows * DRAY;
        float* vb  = kb + (size_t)rows * DRAY;
        float* cat = vb + (size_t)rows * DRAY;
        float* h1  = cat + (size_t)rows * 2 * DRAY;
        run_encoder(stream, xr, RNUM, LRAY, DRAY, 7, rt, qb, kb, vb, cat, h1);
    }

    // ---- stage 4: density MLP [56 -> 32 -> 16 -> 1] -> srdf output ----
    {
        float* hb  = pool;                         // 32768 x 32
        float* h2b = hb + (size_t)NPTS * 32;       // 32768 x 16
        launch_gemm(stream, xr, DRAY, (const float*)d_in[IN_D_W0], DRAY,
                    (const float*)d_in[IN_D_B0], hb, 32, NPTS, 32, DRAY, 1);
        launch_gemm(stream, hb, 32, (const float*)d_in[IN_D_W1], 32,
                    (const float*)d_in[IN_D_B1], h2b, 16, NPTS, 16, 32, 1);
        launch_gemm(stream, h2b, 16, (const float*)d_in[IN_D_W2], 16,
                    (const float*)d_in[IN_D_B2], out_srdf, 1, NPTS, 1, 16, 0);
    }

    // ---- stage 5: radiance MLP [51 -> 16 -> 8 -> 1] + softmax blend, chunked ----
    {
        const int PTCH = 8192;                     // points per chunk
        const int rows = PTCH * NVIEW;             // 65536
        float* rin = pool;                         // rows x 52 (padded K)
        float* hb  = rin + (size_t)rows * 52;      // rows x 16
        float* h2b = hb + (size_t)rows * 16;       // rows x 8
        float* xwb = h2b + (size_t)rows * 8;       // rows x 1
        for (int c = 0; c < NPTS / PTCH; ++c) {
            const int ptBase = c * PTCH;
            rad_in_kernel<<<rows / 256, 256, 0, stream>>>(tokens, dirrel, rin,
                                                          ptBase * NVIEW, rows);
            launch_gemm(stream, rin, 52, (const float*)d_in[IN_R_W0], 51,
                        (const float*)d_in[IN_R_B0], hb, 16, rows, 16, 51, 1);
            launch_gemm(stream, hb, 16, (const float*)d_in[IN_R_W1], 16,
                        (const float*)d_in[IN_R_B1], h2b, 8, rows, 8, 16, 1);
            launch_gemm(stream, h2b, 8, (const float*)d_in[IN_R_W2], 8,
                        (const float*)d_in[IN_R_B2], xwb, 1, rows, 1, 8, 0);
            blend_kernel<<<PTCH / 256, 256, 0, stream>>>(xwb, maskb, rgbb, out_rad,
                                                         ptBase, PTCH);
        }
    }
}